// Fusion_Block_s_15728170238235
// MI455X (gfx1250) — compile-verified
//
#include <hip/hip_runtime.h>
#include <math.h>

typedef __attribute__((ext_vector_type(2))) float v2f;
typedef __attribute__((ext_vector_type(8))) float v8f;

#define EPS_BN 1e-5f

// Problem sizes (fixed by the reference)
constexpr int B = 8, L = 1024, O = 512, D = 32, KF = 8, M = 512, P = 512;

// Branchless tanh: prefer the native CDNA5 v_tanh_f32 transcendental.
__device__ __forceinline__ float tanh_fast(float x) {
#if __has_builtin(__builtin_amdgcn_tanhf)
    return __builtin_amdgcn_tanhf(x);
#elif __has_builtin(__builtin_amdgcn_exp2f) && __has_builtin(__builtin_amdgcn_rcpf)
    // tanh(x) = 1 - 2/(exp2(2*log2(e)*x) + 1)   (saturates correctly at +-inf)
    const float e = __builtin_amdgcn_exp2f(x * 2.885390081777927f);
    return fmaf(-2.0f, __builtin_amdgcn_rcpf(e + 1.0f), 1.0f);
#else
    return tanhf(x);
#endif
}

// ---------------------------------------------------------------------------
// Kernel 1: fused  scores -> tanh(BN) -> reduce-over-L -> BN1  producing xr.
// One workgroup per (b,d).  The 537 MB scores tensor never touches memory.
// Each wave: B operands for its 4 o-tiles live in registers; one LDS A-tile
// load feeds 8 WMMAs; next tile is prefetched during the tanh epilogue.
// ---------------------------------------------------------------------------
__global__ __launch_bounds__(256) void k1_attn_fused(
    const float* __restrict__ x,        // [B,L,D]
    const float* __restrict__ xdate,    // [B,L,D,KF]
    const float* __restrict__ ydate,    // [B,O,D,KF]
    const float* __restrict__ gng, const float* __restrict__ gnb,
    const float* __restrict__ gnm, const float* __restrict__ gnv,
    const float* __restrict__ n1g, const float* __restrict__ n1b,
    const float* __restrict__ n1m, const float* __restrict__ n1v,
    float* __restrict__ xr)             // [B,D,P]
{
    __shared__ float sxd[L * KF];   // 32 KB : x_date[b,:,d,:]
    __shared__ float syd[O * KF];   // 16 KB : y_date[b,:,d,:]
    __shared__ float sx [L];        //  4 KB : x[b,:,d]

    const int d   = blockIdx.x;
    const int b   = blockIdx.y;
    const int tid = threadIdx.x;

    // ---- stage slices into LDS (float4 global loads, contiguous k) ----
    const float* gxd = xdate + ((size_t)b * L * D + d) * KF;   // (l,kk) at l*D*KF + kk
    for (int i = tid; i < L * 2; i += 256) {
        const int l = i >> 1, h = (i & 1) * 4;
        *(float4*)(&sxd[l * KF + h]) = *(const float4*)(gxd + (size_t)l * D * KF + h);
    }
    const float* gyd = ydate + ((size_t)b * O * D + d) * KF;
    for (int i = tid; i < O * 2; i += 256) {
        const int o = i >> 1, h = (i & 1) * 4;
        *(float4*)(&syd[o * KF + h]) = *(const float4*)(gyd + (size_t)o * D * KF + h);
    }
    const float* gx = x + (size_t)b * L * D + d;
    for (int l = tid; l < L; l += 256) sx[l] = gx[(size_t)l * D];
    __syncthreads();

    // ---- per-channel BN constants (d fixed per block -> scalars) ----
    const float gInv  = gng[d] * rsqrtf(gnv[d] + EPS_BN);
    const float gSh   = gnb[d] - gnm[d] * gInv;
    const float n1Inv = n1g[d] * rsqrtf(n1v[d] + EPS_BN);
    const float n1Sh  = n1b[d] - n1m[d] * n1Inv;

    const int lane = tid & 31;
    const int wave = tid >> 5;
    const int n    = lane & 15;      // column / row-in-tile index
    const int hi   = lane >> 4;      // half-wave select
    const int kb   = hi * 2;         // lanes 16-31 hold K=2,3 (ISA 7.12.2)

    // B operands for this wave's 4 o-tiles: register resident for whole L loop
    v2f   bOp1[4], bOp2[4];
    float vAcc[4] = {0.f, 0.f, 0.f, 0.f};
#pragma unroll
    for (int j = 0; j < 4; ++j) {
        const int ob = (wave * 4 + j) * 16;
        bOp1[j] = *(const v2f*)(&syd[(ob + n) * KF + kb]);      // K=0..3
        bOp2[j] = *(const v2f*)(&syd[(ob + n) * KF + kb + 4]);  // K=4..7
    }

    // prefetch l-tile 0
    v2f    a1  = *(const v2f*)(&sxd[n * KF + kb]);
    v2f    a2  = *(const v2f*)(&sxd[n * KF + kb + 4]);
    float4 xlo = *(const float4*)(&sx[hi * 8]);
    float4 xhi = *(const float4*)(&sx[hi * 8 + 4]);

    for (int lt = 0; lt < L / 16; ++lt) {
        const v2f ua1 = a1, ua2 = a2;
        float xv[8];
        *(float4*)&xv[0] = xlo;
        *(float4*)&xv[4] = xhi;

        // branchless prefetch of next l-tile (wraps on last iteration)
        const int lnb = ((lt + 1) & (L / 16 - 1)) * 16;
        a1  = *(const v2f*)(&sxd[(lnb + n) * KF + kb]);
        a2  = *(const v2f*)(&sxd[(lnb + n) * KF + kb + 4]);
        xlo = *(const float4*)(&sx[lnb + hi * 8]);
        xhi = *(const float4*)(&sx[lnb + hi * 8 + 4]);

#pragma unroll
        for (int j = 0; j < 4; ++j) {
            v8f c = {0.f, 0.f, 0.f, 0.f, 0.f, 0.f, 0.f, 0.f};
            c = __builtin_amdgcn_wmma_f32_16x16x4_f32(false, ua1, false, bOp1[j],
                                                      (short)0, c, false, false);
            c = __builtin_amdgcn_wmma_f32_16x16x4_f32(false, ua2, false, bOp2[j],
                                                      (short)0, c, false, false);
            // rows of this tile: m = r + 8*hi  ->  x[lb + m]
#pragma unroll
            for (int r = 0; r < 8; ++r)
                vAcc[j] = fmaf(tanh_fast(fmaf(c[r], gInv, gSh)), xv[r], vAcc[j]);
        }
    }

#pragma unroll
    for (int j = 0; j < 4; ++j) {
        // lanes (n) and (n+16) hold partial sums over m=0..7 / m=8..15
        const float v = vAcc[j] + __shfl_xor(vAcc[j], 16, 32);
        if (hi == 0)
            xr[((size_t)b * D + d) * P + (wave * 4 + j) * 16 + n] =
                fmaf(v, n1Inv, n1Sh);
    }
}

// ---------------------------------------------------------------------------
// Kernel 2a: h = GELU(W1 @ xr + b1)    [B,M,P], one wave per 16x16 tile
// ---------------------------------------------------------------------------
__global__ __launch_bounds__(256) void k2_ffn1(
    const float* __restrict__ w1,   // [M,D]
    const float* __restrict__ b1v,  // [M]
    const float* __restrict__ xr,   // [B,D,P]
    float* __restrict__ h)          // [B,M,P]
{
    const int lane = threadIdx.x & 31, wave = threadIdx.x >> 5;
    const int tile = blockIdx.x * 8 + wave;          // B*(M/16)*(P/16) = 8192
    const int b  = tile >> 10;
    const int mt = (tile >> 5) & 31, pt = tile & 31;
    const int n = lane & 15, hi = lane >> 4, kb = hi * 2;
    const int mb = mt * 16, pb = pt * 16;

    const float* wrow = w1 + (size_t)(mb + n) * D;   // this lane's A row
    const float* xrb  = xr + (size_t)b * D * P;

    v8f c = {0.f, 0.f, 0.f, 0.f, 0.f, 0.f, 0.f, 0.f};
#pragma unroll
    for (int kp = 0; kp < D / 4; ++kp) {
        const int k0 = kp * 4;
        const v2f a = *(const v2f*)(wrow + k0 + kb);
        v2f bOp;
        bOp.x = xrb[(size_t)(k0 + kb) * P + pb + n];
        bOp.y = xrb[(size_t)(k0 + kb + 1) * P + pb + n];
        c = __builtin_amdgcn_wmma_f32_16x16x4_f32(false, a, false, bOp,
                                                  (short)0, c, false, false);
    }
    float* hout = h + ((size_t)b * M + mb) * P + pb + n;
#pragma unroll
    for (int r = 0; r < 8; ++r) {
        const int m = mb + r + 8 * hi;
        float v = c[r] + b1v[m];
        v = 0.5f * v * (1.0f + erff(v * 0.70710678118654752f));   // exact GELU
        hout[(size_t)(r + 8 * hi) * P] = v;
    }
}

// ---------------------------------------------------------------------------
// Kernel 2b: out = BN2(xr + W2 @ h + b2) transposed to [B,P,D]
// ---------------------------------------------------------------------------
__global__ __launch_bounds__(256) void k2_ffn2(
    const float* __restrict__ w2,   // [D,M]
    const float* __restrict__ b2v,  // [D]
    const float* __restrict__ h,    // [B,M,P]
    const float* __restrict__ xr,   // [B,D,P]
    const float* __restrict__ n2g, const float* __restrict__ n2b,
    const float* __restrict__ n2m, const float* __restrict__ n2v,
    float* __restrict__ out)        // [B,P,D]
{
    const int lane = threadIdx.x & 31, wave = threadIdx.x >> 5;
    const int tile = blockIdx.x * 8 + wave;          // B*(D/16)*(P/16) = 512
    const int b  = tile >> 6;
    const int dt = (tile >> 5) & 1, pt = tile & 31;
    const int n = lane & 15, hi = lane >> 4, kb = hi * 2;
    const int db = dt * 16, pb = pt * 16;

    const float* wrow = w2 + (size_t)(db + n) * M;
    const float* hb   = h + (size_t)b * M * P;

    v8f c = {0.f, 0.f, 0.f, 0.f, 0.f, 0.f, 0.f, 0.f};
#pragma unroll 4
    for (int kp = 0; kp < M / 4; ++kp) {
        const int k0 = kp * 4;
        const v2f a = *(const v2f*)(wrow + k0 + kb);
        v2f bOp;
        bOp.x = hb[(size_t)(k0 + kb) * P + pb + n];
        bOp.y = hb[(size_t)(k0 + kb + 1) * P + pb + n];
        c = __builtin_amdgcn_wmma_f32_16x16x4_f32(false, a, false, bOp,
                                                  (short)0, c, false, false);
    }
#pragma unroll
    for (int r = 0; r < 8; ++r) {
        const int dd = db + r + 8 * hi;
        const float inv = n2g[dd] * rsqrtf(n2v[dd] + EPS_BN);
        const float sh  = n2b[dd] - n2m[dd] * inv;
        const float v = c[r] + b2v[dd] + xr[((size_t)b * D + dd) * P + pb + n];
        out[((size_t)b * P + pb + n) * D + dd] = fmaf(v, inv, sh);
    }
}

// ---------------------------------------------------------------------------
extern "C" void kernel_launch(void* const* d_in, const int* in_sizes, int n_in,
                              void* d_out, int out_size, void* d_ws, size_t ws_size,
                              hipStream_t stream) {
    const float* x      = (const float*)d_in[0];
    const float* xdate  = (const float*)d_in[1];
    const float* ydate  = (const float*)d_in[2];
    const float* gng    = (const float*)d_in[3];
    const float* gnb    = (const float*)d_in[4];
    const float* gnm    = (const float*)d_in[5];
    const float* gnv    = (const float*)d_in[6];
    const float* n1g    = (const float*)d_in[7];
    const float* n1b    = (const float*)d_in[8];
    const float* n1m    = (const float*)d_in[9];
    const float* n1v    = (const float*)d_in[10];
    const float* w1     = (const float*)d_in[11];
    const float* b1v    = (const float*)d_in[12];
    const float* w2     = (const float*)d_in[13];
    const float* b2v    = (const float*)d_in[14];
    const float* n2g    = (const float*)d_in[15];
    const float* n2b    = (const float*)d_in[16];
    const float* n2m    = (const float*)d_in[17];
    const float* n2v    = (const float*)d_in[18];

    float* xr = (float*)d_ws;                       // [B,D,P]  0.5 MB
    float* h  = xr + (size_t)B * D * P;             // [B,M,P]  8 MB

    dim3 g1(D, B);
    k1_attn_fused<<<g1, 256, 0, stream>>>(x, xdate, ydate,
                                          gng, gnb, gnm, gnv,
                                          n1g, n1b, n1m, n1v, xr);

    k2_ffn1<<<(B * (M / 16) * (P / 16)) / 8, 256, 0, stream>>>(w1, b1v, xr, h);

    k2_ffn2<<<(B * (D / 16) * (P / 16)) / 8, 256, 0, stream>>>(
        w2, b2v, h, xr, n2g, n2b, n2m, n2v, (float*)d_out);
}